// FocalLoss_40870908788865
// MI455X (gfx1250) — compile-verified
//
#include <hip/hip_runtime.h>

// ---------------------------------------------------------------------------
// Focal loss (gamma=2, w0=w1=1) mean-reduction for MI455X (gfx1250, wave32).
//
// Roofline: 268 MB read @ 23.3 TB/s -> ~11.5 us floor; memory bound.
// Strategy: b128 non-temporal streaming loads, 1 log per element via the
// algebraic identity  per_el = log(a)*(1-a)^2  with  a = t ? p : 1-p,
// f32 accumulation, wave reduction on the matrix pipe (v_wmma_f32_16x16x4_f32),
// deterministic two-kernel partial-sum reduction (no atomics).
// ---------------------------------------------------------------------------

typedef __attribute__((ext_vector_type(2))) float v2f;
typedef __attribute__((ext_vector_type(4))) float v4f;
typedef __attribute__((ext_vector_type(4))) int   v4i;
typedef __attribute__((ext_vector_type(8))) float v8f;

#define FL_BLOCK      256      // 8 wave32 waves per workgroup
#define FL_NUM_BLOCKS 1024     // partials fit in 4 KB of d_ws

// Sum a float across all 32 lanes of a wave.
// WMMA path: A(16x4 f32) carries the 32 lane partials (VGPR0 = partial,
// VGPR1 = 0 -> A[m,0]=partial[lane m], A[m,2]=partial[lane m+16], K=1,3 zero).
// B = ones(4x16). D[m,n] = partial[m] + partial[m+16] = r_m, laid out so that
// lane L's 8 D VGPRs hold r_0..r_7 (lanes 0-15) or r_8..r_15 (lanes 16-31).
// In-lane sum of the 8 VGPRs + one xor-16 shuffle = full wave sum, all in f32.
__device__ __forceinline__ float wave_sum32(float x) {
#if __has_builtin(__builtin_amdgcn_wmma_f32_16x16x4_f32)
    v2f a; a[0] = x;    a[1] = 0.0f;
    v2f b; b[0] = 1.0f; b[1] = 1.0f;
    v8f c = {};
    c = __builtin_amdgcn_wmma_f32_16x16x4_f32(
            /*neg_a=*/false, a, /*neg_b=*/false, b,
            /*c_mod=*/(short)0, c, /*reuse_a=*/false, /*reuse_b=*/false);
    float s = ((c[0] + c[1]) + (c[2] + c[3])) + ((c[4] + c[5]) + (c[6] + c[7]));
    s += __shfl_xor(s, 16, 32);
    return s;
#else
    #pragma unroll
    for (int off = 16; off > 0; off >>= 1) x += __shfl_xor(x, off, 32);
    return x;
#endif
}

__device__ __forceinline__ float focal_elem(float p, int t) {
    // t==1:  log(p)  *(1-p)^2 ; t==0:  log(1-p)*p^2
    float a = (t == 1) ? p : (1.0f - p);
    float q = 1.0f - a;
    return __logf(a) * q * q;   // v_log_f32 + v_mul
}

__global__ void __launch_bounds__(FL_BLOCK)
focal_partial_kernel(const float* __restrict__ inp,
                     const int*   __restrict__ tgt,
                     float*       __restrict__ partial,
                     int n4) {
    const v4f* __restrict__ in4 = (const v4f*)inp;
    const v4i* __restrict__ tg4 = (const v4i*)tgt;

    int tid    = blockIdx.x * FL_BLOCK + threadIdx.x;
    int stride = gridDim.x * FL_BLOCK;

    float acc = 0.0f;
    for (int i = tid; i < n4; i += stride) {
        // Non-temporal b128 loads: 268 MB stream > 192 MB L2, never reused.
        v4f p = __builtin_nontemporal_load(&in4[i]);
        v4i t = __builtin_nontemporal_load(&tg4[i]);
        acc += focal_elem(p[0], t[0]);
        acc += focal_elem(p[1], t[1]);
        acc += focal_elem(p[2], t[2]);
        acc += focal_elem(p[3], t[3]);
    }

    // All 256 threads reach here with EXEC all-ones (WMMA requirement).
    float wsum = wave_sum32(acc);

    __shared__ float lds[FL_BLOCK / 32];
    int lane = threadIdx.x & 31;
    int wid  = threadIdx.x >> 5;
    if (lane == 0) lds[wid] = wsum;
    __syncthreads();

    if (threadIdx.x == 0) {
        float s = 0.0f;
        #pragma unroll
        for (int i = 0; i < FL_BLOCK / 32; ++i) s += lds[i];
        partial[blockIdx.x] = s;
    }
}

__global__ void __launch_bounds__(FL_BLOCK)
focal_final_kernel(const float* __restrict__ partial,
                   float*       __restrict__ out,
                   int nPartials, float invN) {
    float acc = 0.0f;
    for (int i = threadIdx.x; i < nPartials; i += FL_BLOCK)
        acc += partial[i];

    float wsum = wave_sum32(acc);

    __shared__ float lds[FL_BLOCK / 32];
    int lane = threadIdx.x & 31;
    int wid  = threadIdx.x >> 5;
    if (lane == 0) lds[wid] = wsum;
    __syncthreads();

    if (threadIdx.x == 0) {
        float s = 0.0f;
        #pragma unroll
        for (int i = 0; i < FL_BLOCK / 32; ++i) s += lds[i];
        out[0] = -s * invN;     // mean with negation
    }
}

extern "C" void kernel_launch(void* const* d_in, const int* in_sizes, int n_in,
                              void* d_out, int out_size, void* d_ws, size_t ws_size,
                              hipStream_t stream) {
    const float* inp = (const float*)d_in[0];   // probabilities, fp32
    const int*   tgt = (const int*)d_in[1];     // targets {0,1}, int32
    float* partial   = (float*)d_ws;            // FL_NUM_BLOCKS floats
    float* out       = (float*)d_out;           // single fp32 scalar

    int n  = in_sizes[0];       // 33554432, divisible by 4
    int n4 = n >> 2;
    float invN = 1.0f / (float)n;

    focal_partial_kernel<<<FL_NUM_BLOCKS, FL_BLOCK, 0, stream>>>(inp, tgt, partial, n4);
    focal_final_kernel<<<1, FL_BLOCK, 0, stream>>>(partial, out, FL_NUM_BLOCKS, invN);
}